// Head_429496730283
// MI455X (gfx1250) — compile-verified
//
#include <hip/hip_runtime.h>
#include <hip/hip_bf16.h>
#include <math.h>

typedef float v2f __attribute__((ext_vector_type(2)));
typedef float v8f __attribute__((ext_vector_type(8)));

#define Bsz 512
#define Tsz 256
#define Csz 192
#define Hsz 64
#define QS_STR 66    // LDS row stride (floats) for q/k/v tiles: 66*4B -> 8B aligned, bank-spread
#define S_STR  258   // LDS row stride (floats) for score tile

static __device__ __forceinline__ v8f wmma4(v2f a, v2f b, v8f c) {
  // D = A(16x4,f32) * B(4x16,f32) + C(16x16,f32)
  return __builtin_amdgcn_wmma_f32_16x16x4_f32(
      /*neg_a=*/false, a, /*neg_b=*/false, b,
      /*c_mod=*/(short)0, c, /*reuse_a=*/false, /*reuse_b=*/false);
}

// ---------------------------------------------------------------------------
// pos_k = pos_table @ Wk ; pos_q = pos_table @ Wq     (256x192 @ 192x64)
// grid (4,2), block 128 (4 waves). Wave -> 16 rows x 64 cols, K loop of 4.
// ---------------------------------------------------------------------------
__global__ __launch_bounds__(128) void pos_proj_kernel(
    const float* __restrict__ pt, const float* __restrict__ Wk,
    const float* __restrict__ Wq, float* __restrict__ pos_k,
    float* __restrict__ pos_q) {
  const int lane = threadIdx.x & 31, wave = threadIdx.x >> 5;
  const int lm = lane & 15, hi = lane >> 4;
  const float* W = (blockIdx.y == 0) ? Wk : Wq;
  float* O = (blockIdx.y == 0) ? pos_k : pos_q;
  const int m0 = blockIdx.x * 64 + wave * 16;

  v8f acc[4] = {};
  for (int k0 = 0; k0 < Csz; k0 += 4) {
    const int kk = k0 + 2 * hi;
    v2f a = *(const v2f*)(pt + (m0 + lm) * Csz + kk);   // A row-major pair
    for (int nt = 0; nt < 4; ++nt) {
      v2f b;
      b.x = W[kk * Hsz + nt * 16 + lm];
      b.y = W[(kk + 1) * Hsz + nt * 16 + lm];
      acc[nt] = wmma4(a, b, acc[nt]);
    }
  }
  for (int nt = 0; nt < 4; ++nt)
    for (int e = 0; e < 8; ++e)
      O[(m0 + e + 8 * hi) * Hsz + nt * 16 + lm] = acc[nt][e];
}

// ---------------------------------------------------------------------------
// bias = pos_k @ pos_q^T   (256x64 @ 64x256) -> 256x256
// grid 16, block 256 (8 waves). NT GEMM: both fragments are float2 row loads.
// ---------------------------------------------------------------------------
__global__ __launch_bounds__(256) void bias_kernel(
    const float* __restrict__ pk, const float* __restrict__ pq,
    float* __restrict__ bias) {
  const int lane = threadIdx.x & 31, wave = threadIdx.x >> 5;
  const int lm = lane & 15, hi = lane >> 4;
  const int m0 = blockIdx.x * 16;
  for (int nt = wave; nt < 16; nt += 8) {
    v8f acc = {};
    for (int k0 = 0; k0 < Hsz; k0 += 4) {
      const int kk = k0 + 2 * hi;
      v2f a = *(const v2f*)(pk + (m0 + lm) * Hsz + kk);
      v2f b = *(const v2f*)(pq + (nt * 16 + lm) * Hsz + kk);  // B^T pair
      acc = wmma4(a, b, acc);
    }
    for (int e = 0; e < 8; ++e)
      bias[(m0 + e + 8 * hi) * Tsz + nt * 16 + lm] = acc[e];
  }
}

// ---------------------------------------------------------------------------
// Fused per-batch attention. One workgroup per batch (512 blocks, 8 waves).
// LDS: q/k/v (3 x 256x66 f32) + score tile (64x258 f32) + row stats = 269 KB.
// ---------------------------------------------------------------------------
__global__ __launch_bounds__(256) void attn_kernel(
    const float* __restrict__ x, const float* __restrict__ Wk,
    const float* __restrict__ Wq, const float* __restrict__ Wv,
    const float* __restrict__ bias, float* __restrict__ out) {
  __shared__ __align__(16) float qs[Tsz * QS_STR];
  __shared__ __align__(16) float ks[Tsz * QS_STR];
  __shared__ __align__(16) float vs[Tsz * QS_STR];
  __shared__ __align__(16) float Sl[64 * S_STR];
  __shared__ float rinv[64];

  const int b = blockIdx.x;
  const int lane = threadIdx.x & 31, wave = threadIdx.x >> 5;
  const int lm = lane & 15, hi = lane >> 4;
  const float* xb = x + (size_t)b * Tsz * Csz;
  const float scale = 0.07216878365f;  // 192^-0.5

  // ---- Phase A: q,k,v = x[b] @ {Wq,Wk,Wv} into LDS (192 tiles of 16x16) ----
  for (int t = wave; t < 192; t += 8) {
    const int which = t >> 6;          // 0:q 1:k 2:v
    const int rem = t & 63;
    const int mt = rem >> 2, nt = rem & 3;
    const float* W = (which == 0) ? Wq : (which == 1) ? Wk : Wv;
    float* dst = (which == 0) ? qs : (which == 1) ? ks : vs;
    v8f acc = {};
    const int arow = mt * 16 + lm;
    for (int k0 = 0; k0 < Csz; k0 += 4) {
      const int kk = k0 + 2 * hi;
      v2f a = *(const v2f*)(xb + arow * Csz + kk);
      v2f bb;
      bb.x = W[kk * Hsz + nt * 16 + lm];
      bb.y = W[(kk + 1) * Hsz + nt * 16 + lm];
      acc = wmma4(a, bb, acc);
    }
    for (int e = 0; e < 8; ++e)
      dst[(mt * 16 + e + 8 * hi) * QS_STR + nt * 16 + lm] = acc[e];
  }
  __syncthreads();

  // ---- Phase B: per 64-row query tile ----
  for (int qt = 0; qt < 4; ++qt) {
    const int ntiles = (qt + 1) * 4;      // causal: only lower-left key tiles
    const int kmax = (qt + 1) * 64;

    // scores S = scale * q K^T + bias, causal-masked (NT, K=64)
    for (int t = wave; t < 4 * ntiles; t += 8) {
      const int mt = t & 3, nt = t >> 2;
      v8f acc = {};
      const int qrow = qt * 64 + mt * 16 + lm;
      const int krow = nt * 16 + lm;
      for (int k0 = 0; k0 < Hsz; k0 += 4) {
        const int kk = k0 + 2 * hi;
        v2f a = *(const v2f*)(&qs[qrow * QS_STR + kk]);
        v2f bb = *(const v2f*)(&ks[krow * QS_STR + kk]);
        acc = wmma4(a, bb, acc);
      }
      const int kj = nt * 16 + lm;
      for (int e = 0; e < 8; ++e) {
        const int qi = qt * 64 + mt * 16 + e + 8 * hi;
        float v = acc[e] * scale + bias[qi * Tsz + kj];
        if (kj > qi) v = -__builtin_inff();
        Sl[(mt * 16 + e + 8 * hi) * S_STR + kj] = v;
      }
    }
    __syncthreads();

    // row softmax over [0, kmax): wave w owns rows 8w..8w+7
    for (int r = wave * 8; r < wave * 8 + 8; ++r) {
      float m = -__builtin_inff();
      for (int c = lane; c < kmax; c += 32) m = fmaxf(m, Sl[r * S_STR + c]);
      for (int off = 16; off; off >>= 1) m = fmaxf(m, __shfl_xor(m, off, 32));
      float s = 0.f;
      for (int c = lane; c < kmax; c += 32) {
        float e = __expf(Sl[r * S_STR + c] - m);
        Sl[r * S_STR + c] = e;
        s += e;
      }
      for (int off = 16; off; off >>= 1) s += __shfl_xor(s, off, 32);
      if (lane == 0) rinv[r] = 1.f / s;
    }
    __syncthreads();

    // O = P @ V (NN, K=kmax); 16 output tiles, 2 per wave sharing A fragment
    for (int ti = 0; ti < 2; ++ti) {
      const int t = wave + ti * 8;
      const int mt = t & 3, nt = t >> 2;
      v8f acc = {};
      const int prow = mt * 16 + lm;
      const int n = nt * 16 + lm;
      for (int k0 = 0; k0 < kmax; k0 += 4) {
        const int kk = k0 + 2 * hi;
        v2f a = *(const v2f*)(&Sl[prow * S_STR + kk]);
        v2f bb;
        bb.x = vs[kk * QS_STR + n];
        bb.y = vs[(kk + 1) * QS_STR + n];
        acc = wmma4(a, bb, acc);
      }
      for (int e = 0; e < 8; ++e) {
        const int rl = mt * 16 + e + 8 * hi;
        out[((size_t)b * Tsz + qt * 64 + rl) * Hsz + n] = acc[e] * rinv[rl];
      }
    }
    __syncthreads();  // before next qt overwrites Sl
  }
}

// ---------------------------------------------------------------------------
extern "C" void kernel_launch(void* const* d_in, const int* in_sizes, int n_in,
                              void* d_out, int out_size, void* d_ws,
                              size_t ws_size, hipStream_t stream) {
  (void)in_sizes; (void)n_in; (void)out_size; (void)ws_size;
  const float* x   = (const float*)d_in[0];
  const float* Wk  = (const float*)d_in[1];
  const float* Wq  = (const float*)d_in[2];
  const float* Wv  = (const float*)d_in[3];
  const float* pt  = (const float*)d_in[4];
  float* out = (float*)d_out;

  float* pos_k = (float*)d_ws;                 // 256*64 floats
  float* pos_q = pos_k + Tsz * Hsz;            // 256*64 floats
  float* bias  = pos_q + Tsz * Hsz;            // 256*256 floats (384 KB total ws)

  pos_proj_kernel<<<dim3(4, 2), 128, 0, stream>>>(pt, Wk, Wq, pos_k, pos_q);
  bias_kernel<<<16, 256, 0, stream>>>(pos_k, pos_q, bias);
  attn_kernel<<<Bsz, 256, 0, stream>>>(x, Wk, Wq, Wv, bias, out);
}